// EulerScheduler_21784074125653
// MI455X (gfx1250) — compile-verified
//
#include <hip/hip_runtime.h>
#include <hip/hip_bf16.h>
#include <cfloat>

// Problem constants (match reference)
#define B_      2
#define L_      2048
#define V_      32001
#define EPS_F   1e-3f
#define GEPS_F  1e-6f

// Tiling
#define THREADS     256          // 8 wave32 per workgroup
#define TILE        1024         // floats per LDS tile per array
#define PER_THREAD  (TILE / THREADS)          // 4
#define NTILES      ((V_ + TILE - 1) / TILE)  // 32
#define WAVES       (THREADS / 32)

// CDNA5 async global->LDS path (guarded so we always compile)
#ifdef __has_builtin
#if __has_builtin(__builtin_amdgcn_global_load_async_to_lds_b32)
#define HAVE_ASYNC 1
#endif
#if __has_builtin(__builtin_amdgcn_s_wait_asynccnt)
#define HAVE_WAITASYNC_BUILTIN 1
#endif
#endif

#define GLOBAL_AS __attribute__((address_space(1)))
#define LDS_AS    __attribute__((address_space(3)))

template <int N>
__device__ __forceinline__ void wait_asynccnt() {
#if defined(HAVE_ASYNC)
#if defined(HAVE_WAITASYNC_BUILTIN)
    __builtin_amdgcn_s_wait_asynccnt(N);
#else
    asm volatile("s_wait_asynccnt %0" ::"n"(N) : "memory");
#endif
#endif
}

__global__ __launch_bounds__(THREADS) void euler_step_kernel(
    const float* __restrict__ outp,      // [B,L,V] log-score
    const int*   __restrict__ xt,        // [B,L]
    const float* __restrict__ t,         // [B]
    const float* __restrict__ step_size, // [1]
    const float* __restrict__ u,         // [B,L,V]
    float* __restrict__ xnew_out,        // [B,L]   (written as float)
    float* __restrict__ rev_out)         // [B,L,V]
{
    __shared__ float s_o[2][TILE];
    __shared__ float s_u[2][TILE];
    __shared__ float w_sum[WAVES];
    __shared__ float w_max[WAVES];
    __shared__ int   w_idx[WAVES];
    __shared__ float s_gtok;

    const int row = blockIdx.x;          // b*L + l
    const int tid = threadIdx.x;
    const int b   = row / L_;

    int tok = xt[row];
    if (tok < 0 || tok >= V_) tok = V_ - 1;            // xt == -1 -> mask token
    const float msk   = (tok == V_ - 1) ? 1.0f : 0.0f; // is_mask
    const float tb    = t[b];
    const float sigma = (1.0f - EPS_F) / (1.0f - (1.0f - EPS_F) * tb);
    const float dt    = step_size[0];

    const size_t base = (size_t)row * (size_t)V_;
    const float* orow = outp + base;
    const float* urow = u + base;
    float*       rrow = rev_out + base;

    float ssum    = 0.0f;       // sum of score over v != tok
    float best    = -FLT_MAX;   // running max of xt_prob/gnoise (off-diagonal)
    int   bestIdx = 0;

    auto stage = [&](int bufi, int tileBase) {
#pragma unroll
        for (int k = 0; k < PER_THREAD; ++k) {
            int i   = k * THREADS + tid;
            int idx = tileBase + i;
            int c   = (idx < V_) ? idx : (V_ - 1);  // clamp: uniform issue count, no OOB
#if defined(HAVE_ASYNC)
            __builtin_amdgcn_global_load_async_to_lds_b32(
                (GLOBAL_AS int*)(orow + c),
                (LDS_AS int*)(&s_o[bufi][i]), 0, 0);
            __builtin_amdgcn_global_load_async_to_lds_b32(
                (GLOBAL_AS int*)(urow + c),
                (LDS_AS int*)(&s_u[bufi][i]), 0, 0);
#else
            s_o[bufi][i] = __builtin_nontemporal_load(orow + c);
            s_u[bufi][i] = __builtin_nontemporal_load(urow + c);
#endif
        }
    };

    // Software pipeline: prefetch tile 0, then overlap tile t+1 loads with tile t compute.
    stage(0, 0);
    int buf = 0;
#pragma unroll 1
    for (int tile = 0; tile < NTILES; ++tile) {
        const int tileBase = tile * TILE;
        if (tile + 1 < NTILES) {
            stage(buf ^ 1, tileBase + TILE);
            // per-wave: 8 async ops outstanding for the next tile; ASYNCcnt completes
            // in order, so <=8 guarantees the current tile's 8 ops landed in LDS.
            wait_asynccnt<2 * PER_THREAD>();
        } else {
            wait_asynccnt<0>();
        }
        __syncthreads();

#pragma unroll
        for (int k = 0; k < PER_THREAD; ++k) {
            int i   = k * THREADS + tid;
            int idx = tileBase + i;
            if (idx < V_) {
                float o  = s_o[buf][i];
                float uu = s_u[buf][i];
                float sc = __expf(o);
                float g  = GEPS_F - __logf(GEPS_F + (1.0f - GEPS_F) * uu);
                if (idx == tok) {
                    s_gtok = g;  // exactly one thread in the block hits this
                } else {
                    ssum += sc;
                    float rr = msk * (sigma * sc);   // off-diag rev_rate
                    __builtin_nontemporal_store(rr, rrow + idx);
                    float ratio = (dt * rr) / g;     // xt_prob/gnoise, oh==0 off-diag
                    if (ratio > best || (ratio == best && idx < bestIdx)) {
                        best = ratio; bestIdx = idx;
                    }
                }
            }
        }
        __syncthreads();
        buf ^= 1;
    }

    // wave32 reduction: sum + (max, first-index) argmax
#pragma unroll
    for (int off = 16; off > 0; off >>= 1) {
        ssum += __shfl_down(ssum, off, 32);
        float om = __shfl_down(best, off, 32);
        int   oi = __shfl_down(bestIdx, off, 32);
        if (om > best || (om == best && oi < bestIdx)) { best = om; bestIdx = oi; }
    }
    if ((tid & 31) == 0) {
        int w = tid >> 5;
        w_sum[w] = ssum; w_max[w] = best; w_idx[w] = bestIdx;
    }
    __syncthreads();

    if (tid == 0) {
        float S  = 0.0f;
        float bm = -FLT_MAX;
        int   bi = 0;
        for (int w = 0; w < WAVES; ++w) {
            S += w_sum[w];
            if (w_max[w] > bm || (w_max[w] == bm && w_idx[w] < bi)) {
                bm = w_max[w]; bi = w_idx[w];
            }
        }
        // diagonal: r[tok] = -sum_{v!=tok}(is_mask*score); xt_prob[tok] = 1 + dt*rev
        float rr_d    = -(msk * sigma) * S;
        float ratio_d = (1.0f + dt * rr_d) / s_gtok;
        if (ratio_d > bm || (ratio_d == bm && tok < bi)) bi = tok;
        __builtin_nontemporal_store(rr_d, rrow + tok);
        int xn = (bi == V_ - 1) ? -1 : bi;   // map mask token back to -1
        xnew_out[row] = (float)xn;
    }
}

extern "C" void kernel_launch(void* const* d_in, const int* in_sizes, int n_in,
                              void* d_out, int out_size, void* d_ws, size_t ws_size,
                              hipStream_t stream) {
    (void)in_sizes; (void)n_in; (void)out_size; (void)d_ws; (void)ws_size;
    const float* outp = (const float*)d_in[0];
    const int*   xt   = (const int*)d_in[1];
    const float* t    = (const float*)d_in[2];
    const float* step = (const float*)d_in[3];
    const float* u    = (const float*)d_in[4];

    float* xnew = (float*)d_out;                     // B*L entries (x_new as float)
    float* rev  = (float*)d_out + (size_t)B_ * L_;   // B*L*V entries (rev_rate)

    dim3 grid(B_ * L_);
    dim3 block(THREADS);
    euler_step_kernel<<<grid, block, 0, stream>>>(outp, xt, t, step, u, xnew, rev);
}